// GraphNet_51384988729797
// MI455X (gfx1250) — compile-verified
//
#include <hip/hip_runtime.h>
#include <hip/hip_bf16.h>

typedef __attribute__((ext_vector_type(2))) float v2f;
typedef __attribute__((ext_vector_type(8))) float v8f;

#define THREADS 256

static inline unsigned cdivu(long long a, long long b) { return (unsigned)((a + b - 1) / b); }

// ---------------------------------------------------------------- utilities

__global__ void k_zero(float* __restrict__ p, long long n) {
  long long i = (long long)blockIdx.x * blockDim.x + threadIdx.x;
  if (i < n) p[i] = 0.0f;
}

__global__ void k_degree(const int* __restrict__ src, const int* __restrict__ dst,
                         float* __restrict__ outdeg, float* __restrict__ indeg, int E) {
  int e = blockIdx.x * blockDim.x + threadIdx.x;
  if (e < E) {
    atomicAdd(&outdeg[src[e]], 1.0f);
    atomicAdd(&indeg[dst[e]], 1.0f);
  }
}

// deg -> max(deg,1)^-1/2 in place (ns and nd are contiguous: one call covers both)
__global__ void k_invsqrt(float* __restrict__ p, int n) {
  int i = blockIdx.x * blockDim.x + threadIdx.x;
  if (i < n) p[i] = 1.0f / sqrtf(fmaxf(p[i], 1.0f));
}

// C == 1 path: agg[dst[e]] += h[src[e]] * ns[src[e]]
__global__ void k_scatter_add1(const float* __restrict__ h, const float* __restrict__ ns,
                               const int* __restrict__ src, const int* __restrict__ dst,
                               float* __restrict__ agg, int E) {
  int e = blockIdx.x * blockDim.x + threadIdx.x;
  if (e >= E) return;
  int s = src[e];
  atomicAdd(&agg[dst[e]], h[s] * ns[s]);
}

// C = 1<<cshift >= 4: one float4 of channels per thread (global_load_b128 gather,
// 4x global_atomic_add_f32 scatter).
__global__ void k_scatter_add4(const float* __restrict__ h, const float* __restrict__ ns,
                               const int* __restrict__ src, const int* __restrict__ dst,
                               float* __restrict__ agg, long long total4, int cshift) {
  long long i = (long long)blockIdx.x * blockDim.x + threadIdx.x;
  if (i >= total4) return;
  const int vshift = cshift - 2;                 // vec4s per node row
  long long e = i >> vshift;
  int c4 = (int)(i & ((1 << vshift) - 1)) << 2;
  int s = src[e];
  int d = dst[e];
  const float4 v = *(const float4*)(h + ((long long)s << cshift) + c4);
  const float sc = ns[s];
  float* ap = agg + ((long long)d << cshift) + c4;
  atomicAdd(ap + 0, v.x * sc);
  atomicAdd(ap + 1, v.y * sc);
  atomicAdd(ap + 2, v.z * sc);
  atomicAdd(ap + 3, v.w * sc);
}

// h[row][c] = h[row][c] * nd[row] + b[c]
__global__ void k_scale_bias(float* __restrict__ h, const float* __restrict__ nd,
                             const float* __restrict__ b, long long total, int cshift) {
  long long i = (long long)blockIdx.x * blockDim.x + threadIdx.x;
  if (i >= total) return;
  long long row = i >> cshift;
  int c = (int)(i & ((1LL << cshift) - 1));
  h[i] = h[i] * nd[row] + b[c];
}

// out[i] = | out[i] * nd[i] + b[0] |   (LeakyReLU(slope=-1) == abs)
__global__ void k_finalize_abs(float* __restrict__ out, const float* __restrict__ nd,
                               const float* __restrict__ b, int n) {
  int i = blockIdx.x * blockDim.x + threadIdx.x;
  if (i < n) out[i] = fabsf(out[i] * nd[i] + b[0]);
}

// ---------------------------------------------------------------- WMMA GEMM
// C[nrows x FO] = diag(rowscale) * A[nrows x K] @ W[K x FO] + bias
// REQUIRES nrows % 16 == 0 (true here: N = 100000): no per-row guards anywhere,
// so loads/stores stay exec-uniform and WMMA sees EXEC == all-1s.
// One wave32 per 16x16 output tile; 4 waves per block share a 16-column W tile
// staged transposed in LDS (BT[n][k], zero-padded to KS, 8B-aligned pairs).
//
// V_WMMA_F32_16X16X4_F32 fragments (wave32):
//   A 16x4 (2 VGPR): lanes 0-15 rows 0-15 @ K=k0+0,k0+1; lanes 16-31 @ K=k0+2,k0+3
//   B 4x16 (2 VGPR): v0 = K=k0 / k0+2 per lane half; v1 = K=k0+1 / k0+3
//   C/D (8 VGPR):    vr = row r (lanes 0-15) / row r+8 (lanes 16-31)
template <int K, int FO>
__global__ void k_gemm_wmma(const float* __restrict__ A, const float* __restrict__ rowscale,
                            const float* __restrict__ W, const float* __restrict__ bias,
                            float* __restrict__ C, int nrows) {
  constexpr int KS = (K >= 4) ? K : 4;            // LDS row stride (zero-padded K)
  __shared__ float BT[16 * KS];                   // BT[n*KS + k] = W[k][col0+n]

  const int lane = threadIdx.x;                   // 0..31
  const int wave = threadIdx.y;                   // 0..3
  const int col0 = blockIdx.y * 16;

  for (int idx = wave * 32 + lane; idx < 16 * KS; idx += 128) {
    const int n = idx / KS;
    const int k = idx - n * KS;
    float wv = 0.0f;
    if ((K == KS || k < K) && (FO % 16 == 0 || (col0 + n) < FO))
      wv = W[k * FO + col0 + n];
    BT[idx] = wv;
  }
  __syncthreads();

  const int rowTile = blockIdx.x * blockDim.y + wave;
  const int row0 = rowTile * 16;
  if (row0 >= nrows) return;                      // wave-uniform
  const int half = lane >> 4;
  const int l15 = lane & 15;
  const int rA = row0 + l15;                      // always < nrows (nrows%16==0)
  const float scale = rowscale ? rowscale[rA] : 1.0f;

  const float* __restrict__ Arow = A + (long long)rA * K;
  const float* __restrict__ Brow = BT + l15 * KS + half * 2;

  v8f acc = {};
  if constexpr (K >= 4) {
#pragma unroll
    for (int k0 = 0; k0 < K; k0 += 4) {
      v2f a = *(const v2f*)(Arow + k0 + half * 2);      // global_load_b64, no mask
      a.x *= scale;
      a.y *= scale;
      const v2f b = *(const v2f*)(Brow + k0);           // ds_load_b64, no mask
      acc = __builtin_amdgcn_wmma_f32_16x16x4_f32(false, a, false, b, (short)0, acc,
                                                  false, false);
    }
  } else {
    // K == 1: unconditional loads; zero the A-side K>=1 columns via select on the
    // scale (B-side K=2,3 values are zero-padded in BT, so no masking needed there).
    const float am = half ? 0.0f : scale;               // v_cndmask, EXEC untouched
    v2f a;
    a.x = Arow[0] * am;
    a.y = 0.0f;
    const v2f b = *(const v2f*)(Brow);                  // ds_load_b64
    acc = __builtin_amdgcn_wmma_f32_16x16x4_f32(false, a, false, b, (short)0, acc,
                                                false, false);
  }

  const int colC = col0 + l15;
  if (FO % 16 == 0 || colC < FO) {                      // compile-time true unless FO==1
    const float bv = bias ? bias[colC] : 0.0f;
    float* __restrict__ Crow = C + (long long)(row0 + half * 8) * FO + colC;
#pragma unroll
    for (int r = 0; r < 8; ++r)
      Crow[(long long)r * FO] = acc[r] + bv;            // no row guard: tile fully valid
  }
}

template <int K, int FO>
static void launch_gemm(const float* A, const float* rowscale, const float* W,
                        const float* bias, float* C, int nrows, hipStream_t stream) {
  dim3 block(32, 4);                               // 4 waves / block
  dim3 grid(cdivu((nrows + 15) / 16, 4), (unsigned)((FO + 15) / 16));
  k_gemm_wmma<K, FO><<<grid, block, 0, stream>>>(A, rowscale, W, bias, C, nrows);
}

// ---------------------------------------------------------------- driver

extern "C" void kernel_launch(void* const* d_in, const int* in_sizes, int n_in,
                              void* d_out, int out_size, void* d_ws, size_t ws_size,
                              hipStream_t stream) {
  const float* x   = (const float*)d_in[0];
  const int*   src = (const int*)d_in[1];
  const int*   dst = (const int*)d_in[2];
  const float* W[6];
  const float* b[6];
  for (int i = 0; i < 6; ++i) {
    W[i] = (const float*)d_in[3 + 2 * i];
    b[i] = (const float*)d_in[4 + 2 * i];
  }
  const int N = in_sizes[0];     // x is (N,1); N % 16 == 0 for this problem
  const int E = in_sizes[1];

  // workspace layout (floats): ns[N] | nd[N] | bufX[N*128] | bufY[N*128]
  float* ws   = (float*)d_ws;
  float* ns   = ws;
  float* nd   = ws + N;
  float* bufX = nd + N;
  float* bufY = bufX + (size_t)N * 128;
  float* out  = (float*)d_out;

  // ---- degree norms: ns = out_deg^-1/2, nd = in_deg^-1/2 (clamped at 1)
  k_zero<<<cdivu(2LL * N, THREADS), THREADS, 0, stream>>>(ns, 2LL * N);
  k_degree<<<cdivu(E, THREADS), THREADS, 0, stream>>>(src, dst, ns, nd, E);
  k_invsqrt<<<cdivu(2LL * N, THREADS), THREADS, 0, stream>>>(ns, 2 * N);

  // ---- L0: 1 -> 32  (aggregate first, then (agg*nd)@W0 + b0)
  k_zero<<<cdivu(N, THREADS), THREADS, 0, stream>>>(bufY, N);
  k_scatter_add1<<<cdivu(E, THREADS), THREADS, 0, stream>>>(x, ns, src, dst, bufY, E);
  launch_gemm<1, 32>(bufY, nd, W[0], b[0], bufX, N, stream);

  // ---- L1: 32 -> 64
  k_zero<<<cdivu((long long)N * 32, THREADS), THREADS, 0, stream>>>(bufY, (long long)N * 32);
  k_scatter_add4<<<cdivu((long long)E * 8, THREADS), THREADS, 0, stream>>>(
      bufX, ns, src, dst, bufY, (long long)E * 8, 5);
  launch_gemm<32, 64>(bufY, nd, W[1], b[1], bufX, N, stream);

  // ---- L2: 64 -> 128
  k_zero<<<cdivu((long long)N * 64, THREADS), THREADS, 0, stream>>>(bufY, (long long)N * 64);
  k_scatter_add4<<<cdivu((long long)E * 16, THREADS), THREADS, 0, stream>>>(
      bufX, ns, src, dst, bufY, (long long)E * 16, 6);
  launch_gemm<64, 128>(bufY, nd, W[2], b[2], bufX, N, stream);

  // ---- L3: 128 -> 64  (h@W3 first, then aggregate, then *nd + b3)
  launch_gemm<128, 64>(bufX, nullptr, W[3], nullptr, bufY, N, stream);
  k_zero<<<cdivu((long long)N * 64, THREADS), THREADS, 0, stream>>>(bufX, (long long)N * 64);
  k_scatter_add4<<<cdivu((long long)E * 16, THREADS), THREADS, 0, stream>>>(
      bufY, ns, src, dst, bufX, (long long)E * 16, 6);
  k_scale_bias<<<cdivu((long long)N * 64, THREADS), THREADS, 0, stream>>>(
      bufX, nd, b[3], (long long)N * 64, 6);

  // ---- L4: 64 -> 32
  launch_gemm<64, 32>(bufX, nullptr, W[4], nullptr, bufY, N, stream);
  k_zero<<<cdivu((long long)N * 32, THREADS), THREADS, 0, stream>>>(bufX, (long long)N * 32);
  k_scatter_add4<<<cdivu((long long)E * 8, THREADS), THREADS, 0, stream>>>(
      bufY, ns, src, dst, bufX, (long long)E * 8, 5);
  k_scale_bias<<<cdivu((long long)N * 32, THREADS), THREADS, 0, stream>>>(
      bufX, nd, b[4], (long long)N * 32, 5);

  // ---- L5: 32 -> 1, scatter straight into d_out, fused abs epilogue
  launch_gemm<32, 1>(bufX, nullptr, W[5], nullptr, bufY, N, stream);
  k_zero<<<cdivu(N, THREADS), THREADS, 0, stream>>>(out, N);
  k_scatter_add1<<<cdivu(E, THREADS), THREADS, 0, stream>>>(bufY, ns, src, dst, out, E);
  k_finalize_abs<<<cdivu(N, THREADS), THREADS, 0, stream>>>(out, nd, b[5], N);
}